// GNN_38233798869355
// MI455X (gfx1250) — compile-verified
//
#include <hip/hip_runtime.h>

typedef __attribute__((ext_vector_type(2))) float v2f;
typedef __attribute__((ext_vector_type(8))) float v8f;

#define NODES 64
#define EDGES 256
#define EPLUS 320
#define FDIM  64
#define HEADS 4
#define CH    16
#define LD    65          // padded LDS row stride (floats)
#define NEG_SLOPE 0.2f

// Fused per-(graph,sample) GAT: gather -> GAT1 (scalar-feature closed form)
// -> GEMM h1@W2 (f32 WMMA) -> edge softmax -> dense alpha GEMM (f32 WMMA)
// -> bias+relu -> node mean-pool -> dot fc_w.  One 256-thread WG per (g,s).
__global__ __launch_bounds__(256)
void gat_fused_kernel(const float* __restrict__ gene_expr,   // (64, NG)
                      const int*   __restrict__ gene_idx,    // (128, 64)
                      const int*   __restrict__ edge_index,  // (128, 2, 256)
                      const float* __restrict__ W1,          // (64)
                      const float* __restrict__ a1_src,      // (4,16)
                      const float* __restrict__ a1_dst,      // (4,16)
                      const float* __restrict__ b1,          // (64)
                      const float* __restrict__ W2,          // (64,64)
                      const float* __restrict__ a2_src,      // (64)
                      const float* __restrict__ a2_dst,      // (64)
                      const float* __restrict__ b2,          // (64)
                      const float* __restrict__ fc_w,        // (64)
                      const float* __restrict__ fc_b,        // (1)
                      int NG,
                      float* __restrict__ flat_out)          // (64,128): [s*128+g]
{
    const int g    = blockIdx.x;          // graph  0..127
    const int s    = blockIdx.y;          // sample 0..63
    const int tid  = threadIdx.x;         // 0..255
    const int lane = tid & 31;            // wave32
    const int wave = tid >> 5;            // 0..7

    __shared__ float x_lds[NODES];
    __shared__ int   esrc[EPLUS];
    __shared__ int   edst[EPLUS];
    __shared__ float es1[HEADS], ed1[HEADS];
    __shared__ float h1 [NODES][LD];      // layer-1 out, later reused for layer-2 out
    __shared__ float w2p[NODES][LD];      // W2 during GEMM1, reused as dense alpha P
    __shared__ float h2 [NODES][LD];      // h1 @ W2
    __shared__ float e2s[NODES], e2d[NODES];
    __shared__ float red[FDIM];

    // ---------------- stage x, edges, W2, per-head attention scalars ----------------
    if (tid < NODES) {
        int gi = gene_idx[g * NODES + tid];
        x_lds[tid] = gene_expr[s * NG + gi];
    }
    for (int k = tid; k < EPLUS; k += 256) {
        if (k < EDGES) {
            esrc[k] = edge_index[g * 2 * EDGES + k];
            edst[k] = edge_index[g * 2 * EDGES + EDGES + k];
        } else {               // self loops
            esrc[k] = k - EDGES;
            edst[k] = k - EDGES;
        }
    }
    if (tid < HEADS) {
        float ss = 0.f, dd = 0.f;
        for (int c = 0; c < CH; ++c) {
            ss += W1[tid * CH + c] * a1_src[tid * CH + c];
            dd += W1[tid * CH + c] * a1_dst[tid * CH + c];
        }
        es1[tid] = ss; ed1[tid] = dd;
    }
    for (int i = tid; i < FDIM * FDIM; i += 256)
        w2p[i >> 6][i & 63] = W2[i];
    __syncthreads();

    // ---------------- layer 1: per (head,node) scalar softmax-aggregate ----------------
    {
        const int h = tid >> 6;           // 0..3
        const int n = tid & 63;
        const float cs = es1[h];
        const float cd = x_lds[n] * ed1[h];
        float m = -__builtin_inff();
        for (int k = 0; k < EPLUS; ++k)
            if (edst[k] == n) {
                float e = x_lds[esrc[k]] * cs + cd;
                e = (e > 0.f) ? e : NEG_SLOPE * e;
                m = fmaxf(m, e);
            }
        float denom = 0.f, num = 0.f;
        for (int k = 0; k < EPLUS; ++k)
            if (edst[k] == n) {
                float xv = x_lds[esrc[k]];
                float e  = xv * cs + cd;
                e = (e > 0.f) ? e : NEG_SLOPE * e;
                float p = __expf(e - m);
                denom += p;
                num   += xv * p;
            }
        const float A = num / denom;      // self-loop guarantees denom > 0
        for (int c = 0; c < CH; ++c) {
            int f = h * CH + c;
            float v = W1[f] * A + b1[f];
            h1[n][f] = (v > 0.f) ? v : 0.f;
        }
    }
    __syncthreads();

    // ---------------- GEMM1: h2 = h1 @ W2  (64x64x64, V_WMMA_F32_16X16X4_F32) ----------------
    {
        const int mrow  = lane & 15;
        const int khalf = (lane >> 4) << 1;   // 0 or 2
        const int drow0 = (lane >> 4) << 3;   // 0 or 8
        for (int t = wave * 2; t < wave * 2 + 2; ++t) {
            const int mbase = (t >> 2) << 4;
            const int nbase = (t & 3) << 4;
            v8f acc = {};
            for (int ks = 0; ks < 16; ++ks) {
                const int k0 = ks * 4 + khalf;
                v2f a, b;
                a.x = h1[mbase + mrow][k0];
                a.y = h1[mbase + mrow][k0 + 1];
                b.x = w2p[k0    ][nbase + mrow];
                b.y = w2p[k0 + 1][nbase + mrow];
                acc = __builtin_amdgcn_wmma_f32_16x16x4_f32(
                        false, a, false, b, (short)0, acc, false, false);
            }
            for (int i = 0; i < 8; ++i)
                h2[mbase + drow0 + i][nbase + mrow] = acc[i];
        }
    }
    __syncthreads();

    // ---------------- layer-2 attention logits + dense alpha matrix P ----------------
    if (tid < 128) {
        const int n = tid & 63;
        const float* av = (tid < 64) ? a2_src : a2_dst;
        float acc = 0.f;
        for (int f = 0; f < FDIM; ++f) acc += h2[n][f] * av[f];
        if (tid < 64) e2s[n] = acc; else e2d[n] = acc;
    }
    for (int i = tid; i < NODES * LD; i += 256)   // zero P (reuses w2p)
        ((float*)w2p)[i] = 0.f;
    __syncthreads();

    if (tid < NODES) {                    // thread owns dst row n of P
        const int n  = tid;
        const float cd = e2d[n];
        float m = -__builtin_inff();
        for (int k = 0; k < EPLUS; ++k)
            if (edst[k] == n) {
                float e = e2s[esrc[k]] + cd;
                e = (e > 0.f) ? e : NEG_SLOPE * e;
                m = fmaxf(m, e);
            }
        float denom = 0.f;
        for (int k = 0; k < EPLUS; ++k)
            if (edst[k] == n) {
                float e = e2s[esrc[k]] + cd;
                e = (e > 0.f) ? e : NEG_SLOPE * e;
                float p = __expf(e - m);
                denom += p;
                w2p[n][esrc[k]] += p;     // duplicate edges accumulate, as in scatter-add
            }
        const float inv = 1.f / denom;
        for (int j = 0; j < NODES; ++j) w2p[n][j] *= inv;
    }
    __syncthreads();

    // ---------------- GEMM2: out2 = P @ h2 (+b2, relu) -> reuse h1 ----------------
    {
        const int mrow  = lane & 15;
        const int khalf = (lane >> 4) << 1;
        const int drow0 = (lane >> 4) << 3;
        for (int t = wave * 2; t < wave * 2 + 2; ++t) {
            const int mbase = (t >> 2) << 4;
            const int nbase = (t & 3) << 4;
            v8f acc = {};
            for (int ks = 0; ks < 16; ++ks) {
                const int k0 = ks * 4 + khalf;
                v2f a, b;
                a.x = w2p[mbase + mrow][k0];
                a.y = w2p[mbase + mrow][k0 + 1];
                b.x = h2[k0    ][nbase + mrow];
                b.y = h2[k0 + 1][nbase + mrow];
                acc = __builtin_amdgcn_wmma_f32_16x16x4_f32(
                        false, a, false, b, (short)0, acc, false, false);
            }
            const int col = nbase + mrow;
            const float bb = b2[col];
            for (int i = 0; i < 8; ++i) {
                float v = acc[i] + bb;
                h1[mbase + drow0 + i][col] = (v > 0.f) ? v : 0.f;
            }
        }
    }
    __syncthreads();

    // ---------------- mean-pool over nodes, dot fc_w ----------------
    if (tid < FDIM) {
        float ssum = 0.f;
        for (int n = 0; n < NODES; ++n) ssum += h1[n][tid];
        red[tid] = ssum * fc_w[tid];
    }
    __syncthreads();
    if (tid == 0) {
        float r = 0.f;
        for (int f = 0; f < FDIM; ++f) r += red[f];
        flat_out[s * 128 + g] = r * (1.f / 64.f) + fc_b[0];
    }
}

// Tiny readout MLP: (64,128) -> relu(128) -> relu(128) -> 1.  Single block.
__global__ __launch_bounds__(128)
void mlp_kernel(const float* __restrict__ flat,   // (64,128)
                const float* __restrict__ fc1_w, const float* __restrict__ fc1_b,
                const float* __restrict__ fc2_w, const float* __restrict__ fc2_b,
                const float* __restrict__ fc3_w, const float* __restrict__ fc3_b,
                float* __restrict__ out)          // (64)
{
    __shared__ float A [64][128];
    __shared__ float Bf[64][128];
    const int tid = threadIdx.x;      // 0..127
    for (int i = tid; i < 64 * 128; i += 128) A[i >> 7][i & 127] = flat[i];
    __syncthreads();
    {   // layer 1: Bf[s][j] = relu(A[s,:] . fc1_w[:,j] + b)
        const int j = tid;
        const float bj = fc1_b[j];
        for (int ss = 0; ss < 64; ++ss) {
            float acc = bj;
            for (int gg = 0; gg < 128; ++gg) acc += A[ss][gg] * fc1_w[gg * 128 + j];
            Bf[ss][j] = (acc > 0.f) ? acc : 0.f;
        }
    }
    __syncthreads();
    {   // layer 2 -> A
        const int j = tid;
        const float bj = fc2_b[j];
        for (int ss = 0; ss < 64; ++ss) {
            float acc = bj;
            for (int kk = 0; kk < 128; ++kk) acc += Bf[ss][kk] * fc2_w[kk * 128 + j];
            A[ss][j] = (acc > 0.f) ? acc : 0.f;
        }
    }
    __syncthreads();
    if (tid < 64) {   // layer 3
        float acc = fc3_b[0];
        for (int kk = 0; kk < 128; ++kk) acc += A[tid][kk] * fc3_w[kk];
        out[tid] = acc;
    }
}

extern "C" void kernel_launch(void* const* d_in, const int* in_sizes, int n_in,
                              void* d_out, int out_size, void* d_ws, size_t ws_size,
                              hipStream_t stream) {
    const float* gene_expr  = (const float*)d_in[0];
    const int*   gene_idx   = (const int*)  d_in[1];
    const int*   edge_index = (const int*)  d_in[2];
    const float* W1     = (const float*)d_in[3];
    const float* a1_src = (const float*)d_in[4];
    const float* a1_dst = (const float*)d_in[5];
    const float* b1     = (const float*)d_in[6];
    const float* W2     = (const float*)d_in[7];
    const float* a2_src = (const float*)d_in[8];
    const float* a2_dst = (const float*)d_in[9];
    const float* b2     = (const float*)d_in[10];
    const float* fc_w   = (const float*)d_in[11];
    const float* fc_b   = (const float*)d_in[12];
    const float* fc1_w  = (const float*)d_in[13];
    const float* fc1_b  = (const float*)d_in[14];
    const float* fc2_w  = (const float*)d_in[15];
    const float* fc2_b  = (const float*)d_in[16];
    const float* fc3_w  = (const float*)d_in[17];
    const float* fc3_b  = (const float*)d_in[18];

    const int NG = in_sizes[0] / 64;          // 20000
    float* flat = (float*)d_ws;               // 64*128 floats scratch

    dim3 grid(128, 64);                       // (graph, sample)
    gat_fused_kernel<<<grid, 256, 0, stream>>>(
        gene_expr, gene_idx, edge_index, W1, a1_src, a1_dst, b1,
        W2, a2_src, a2_dst, b2, fc_w, fc_b, NG, flat);

    mlp_kernel<<<1, 128, 0, stream>>>(flat, fc1_w, fc1_b, fc2_w, fc2_b,
                                      fc3_w, fc3_b, (float*)d_out);
}